// solvgnn_ternary_40785009443205
// MI455X (gfx1250) — compile-verified
//
#include <hip/hip_runtime.h>
#include <hip/hip_bf16.h>

typedef __attribute__((ext_vector_type(2))) float v2f;
typedef __attribute__((ext_vector_type(8))) float v8f;

#define N_PER 32
#define NB 512
#define NN (N_PER * NB)   // 16384 nodes
#define HD 128

// ---------------------------------------------------------------------------
// fp32 WMMA GEMM: C[M,N] = op(A[M,K] @ B[K,N] + bias).
// Block = 8 waves = 8 consecutive 16-row tiles sharing one 16-col tile.
// The K x 16 B panel is staged into LDS via CDNA5 async global->LDS copies
// (global_load_async_to_lds_b128, tracked by ASYNCcnt), then each wave runs
// the K-loop with V_WMMA_F32_16X16X4_F32 at full fp32 precision.
// Requires: M % 128 == 0, N % 16 == 0, K % 4 == 0, K <= 384.
// ---------------------------------------------------------------------------
__global__ void gemm_wmma_f32(const float* __restrict__ A, int lda,
                              const float* __restrict__ B, int ldb,
                              float* __restrict__ C, int ldc,
                              int M, int N, int K,
                              const float* __restrict__ bias, int relu) {
  __shared__ float Bs[384 * 16];                 // 24 KB max panel
  const int lane = threadIdx.x & 31;
  const int wave = threadIdx.x >> 5;
  const int rowBlks = M >> 7;                    // M / 128
  const int colTile = blockIdx.x / rowBlks;
  const int rowBlk  = blockIdx.x % rowBlks;
  const int row0 = (rowBlk << 7) + (wave << 4);
  const int col0 = colTile << 4;

  // Async-stage the B panel (K rows x 16 cols) into LDS, 16B per transfer.
  for (int i = threadIdx.x; i < (K << 2); i += 256) {
    const int k = i >> 2;
    const int c4 = (i & 3) << 2;
    const unsigned int laddr = (unsigned int)(uintptr_t)(Bs + k * 16 + c4);
    const float* gaddr = B + (size_t)k * ldb + col0 + c4;
    asm volatile("global_load_async_to_lds_b128 %0, %1, off"
                 :: "v"(laddr), "v"(gaddr) : "memory");
  }
  asm volatile("s_wait_asynccnt 0x0" ::: "memory");
  __syncthreads();

  const int h = lane >> 4;                       // half: 0 or 1
  const int l = lane & 15;
  v8f acc = {};
  const float* __restrict__ arow = A + (size_t)(row0 + l) * lda;
  for (int k0 = 0; k0 < K; k0 += 4) {
    const int ka = k0 + 2 * h;
    v2f a, b;
    a.x = arow[ka];
    a.y = arow[ka + 1];
    b.x = Bs[ka * 16 + l];
    b.y = Bs[(ka + 1) * 16 + l];
    acc = __builtin_amdgcn_wmma_f32_16x16x4_f32(
        /*neg_a=*/false, a, /*neg_b=*/false, b,
        /*c_mod=*/(short)0, acc, /*reuse_a=*/false, /*reuse_b=*/false);
  }
  const float bv = bias ? bias[col0 + l] : 0.0f;
  for (int r = 0; r < 8; ++r) {
    float v = acc[r] + bv;
    if (relu) v = fmaxf(v, 0.0f);
    C[(size_t)(row0 + r + 8 * h) * ldc + col0 + l] = v;
  }
}

// ---------------------------------------------------------------------------
// Ring-graph normalized aggregation: Y[v,c] = 0.5*(X[prev,c]+X[next,c]),
// zero-padded out to Cout columns (for K-padding of the 74-wide layer).
// ---------------------------------------------------------------------------
__global__ void ring_agg(const float* __restrict__ X, float* __restrict__ Y,
                         int Cin, int Cout) {
  const int tid = blockIdx.x * blockDim.x + threadIdx.x;
  const int total = NN * Cout;
  if (tid >= total) return;
  const int c = tid % Cout;
  const int v = tid / Cout;
  float val = 0.0f;
  if (c < Cin) {
    const int base = v & ~(N_PER - 1);
    const int loc = v & (N_PER - 1);
    const int p = base + ((loc + N_PER - 1) & (N_PER - 1));
    const int nx = base + ((loc + 1) & (N_PER - 1));
    val = 0.5f * (X[(size_t)p * Cin + c] + X[(size_t)nx * Cin + c]);
  }
  Y[tid] = val;
}

// Zero-pad W1 (74x128) into (80x128)
__global__ void pad_w1(const float* __restrict__ W1, float* __restrict__ W1p) {
  const int tid = blockIdx.x * blockDim.x + threadIdx.x;
  if (tid >= 80 * HD) return;
  const int r = tid >> 7;
  W1p[tid] = (r < 74) ? W1[tid] : 0.0f;
}

// Mean-pool 32 atoms per molecule, scale by solvent fraction, write node row.
__global__ void pool_scale(const float* __restrict__ Y, float* __restrict__ nodes,
                           const float* __restrict__ s1, const float* __restrict__ s2,
                           int mode) {
  const int b = blockIdx.x;    // molecule
  const int o = threadIdx.x;   // 128 channels
  float acc = 0.0f;
  for (int a = 0; a < N_PER; ++a)
    acc += Y[(size_t)(b * N_PER + a) * HD + o];
  acc *= (1.0f / (float)N_PER);
  const float sc = (mode == 0) ? s1[b] : (mode == 1) ? s2[b] : (1.0f - s1[b] - s2[b]);
  nodes[(size_t)(mode * NB + b) * HD + o] = acc * sc;
}

// Generic transpose: WT[c*R + r] = W[r*C + c]
__global__ void transpose_k(const float* __restrict__ W, float* __restrict__ WT,
                            int R, int C) {
  const int tid = blockIdx.x * blockDim.x + threadIdx.x;
  if (tid >= R * C) return;
  const int c = tid % C;
  const int r = tid / C;
  WT[(size_t)c * R + r] = W[tid];
}

// Permute eW2[k, h*128+o] -> M2[h, k*128+o]  (so A-tensor is one big GEMM)
__global__ void ew2_perm(const float* __restrict__ eW2, float* __restrict__ M2) {
  const int tid = blockIdx.x * blockDim.x + threadIdx.x;
  if (tid >= 32 * HD * HD) return;
  const int o = tid & 127;
  const int k = (tid >> 7) & 31;
  const int h = tid >> 12;
  M2[(size_t)h * 4096 + k * HD + o] = eW2[(size_t)k * 16384 + h * HD + o];
}

// Per-destination-node message aggregation over the fixed 9-edge solvent graph:
// xr[v,o] = relu( nn_b[o] + sum_{3 in-edges}( xb[src,o] + sum_k g(hb)[k]*A[src,k,o] ) )
__global__ void edge_agg(const float* __restrict__ Abuf, const float* __restrict__ xb,
                         const float* __restrict__ eW1, const float* __restrict__ eb1,
                         const float* __restrict__ nn_b,
                         const float* __restrict__ hb12, const float* __restrict__ hb13,
                         const float* __restrict__ hb23, const float* __restrict__ hb1,
                         const float* __restrict__ hb2, const float* __restrict__ hb3,
                         float* __restrict__ xr) {
  __shared__ float g3[96];
  const int blk = blockIdx.x;          // 0..3*NB-1
  const int t = blk / NB;
  const int b = blk % NB;
  const int tid = threadIdx.x;         // 128 threads

  float hv0, hv1, hv2;
  int s0, s1n, s2n;
  if (t == 0)      { hv0 = hb12[b]; s0 = NB + b;     hv1 = hb13[b]; s1n = 2*NB + b; hv2 = hb1[b]; s2n = b; }
  else if (t == 1) { hv0 = hb12[b]; s0 = b;          hv1 = hb23[b]; s1n = 2*NB + b; hv2 = hb2[b]; s2n = NB + b; }
  else             { hv0 = hb13[b]; s0 = b;          hv1 = hb23[b]; s1n = NB + b;   hv2 = hb3[b]; s2n = 2*NB + b; }

  if (tid < 96) {
    const int e = tid >> 5, k = tid & 31;
    const float hv = (e == 0) ? hv0 : (e == 1) ? hv1 : hv2;
    g3[tid] = fmaxf(hv * eW1[k] + eb1[k], 0.0f);
  }
  __syncthreads();

  const int o = tid;
  float acc = nn_b[o];
  const int srcs[3] = {s0, s1n, s2n};
  for (int e = 0; e < 3; ++e) {
    const size_t srow = (size_t)srcs[e];
    acc += xb[srow * HD + o];
    const float* __restrict__ Ar = Abuf + srow * 4096 + o;
    for (int k = 0; k < 32; ++k)
      acc += g3[e * 32 + k] * Ar[(size_t)k * HD];
  }
  xr[(size_t)blk * HD + o] = fmaxf(acc, 0.0f);
}

// GRU cell elementwise
__global__ void gru_elem(const float* __restrict__ gi, const float* __restrict__ gh,
                         const float* __restrict__ hdn, float* __restrict__ xnew) {
  const int tid = blockIdx.x * blockDim.x + threadIdx.x;
  if (tid >= 3 * NB * HD) return;
  const int row = tid >> 7;
  const int o = tid & 127;
  const float* gir = gi + (size_t)row * 384;
  const float* ghr = gh + (size_t)row * 384;
  const float r = 1.0f / (1.0f + expf(-(gir[o] + ghr[o])));
  const float z = 1.0f / (1.0f + expf(-(gir[HD + o] + ghr[HD + o])));
  const float n = tanhf(gir[2 * HD + o] + r * ghr[2 * HD + o]);
  xnew[tid] = (1.0f - z) * n + z * hdn[tid];
}

// Concatenate the three node groups per system: hgcat[b, t*128+o]
__global__ void concat3(const float* __restrict__ xnew, float* __restrict__ hgcat) {
  const int tid = blockIdx.x * blockDim.x + threadIdx.x;
  if (tid >= NB * 384) return;
  const int b = tid / 384;
  const int c = tid % 384;
  const int t = c >> 7;
  const int o = c & 127;
  hgcat[tid] = xnew[(size_t)(t * NB + b) * HD + o];
}

// Final 128 -> 3 projection
__global__ void final_out(const float* __restrict__ t2, const float* __restrict__ c3W,
                          const float* __restrict__ c3b, float* __restrict__ out) {
  const int tid = blockIdx.x * blockDim.x + threadIdx.x;
  if (tid >= NB * 3) return;
  const int b = tid / 3;
  const int j = tid % 3;
  float acc = c3b[j];
  const float* r = t2 + (size_t)b * HD;
  for (int o = 0; o < HD; ++o) acc += r[o] * c3W[o * 3 + j];
  out[tid] = acc;
}

// ---------------------------------------------------------------------------
static inline void launch_gemm(const float* A, int lda, const float* B, int ldb,
                               float* C, int ldc, int M, int N, int K,
                               const float* bias, int relu, hipStream_t s) {
  const int blocks = (N / 16) * (M / 128);       // 8 waves/block, 1 tile/wave
  gemm_wmma_f32<<<blocks, 256, 0, s>>>(A, lda, B, ldb, C, ldc, M, N, K, bias, relu);
}

static inline int ceil_div(int a, int b) { return (a + b - 1) / b; }

extern "C" void kernel_launch(void* const* d_in, const int* in_sizes, int n_in,
                              void* d_out, int out_size, void* d_ws, size_t ws_size,
                              hipStream_t stream) {
  const float* h1     = (const float*)d_in[0];
  const float* h2     = (const float*)d_in[1];
  const float* h3     = (const float*)d_in[2];
  const float* solv1  = (const float*)d_in[3];
  const float* solv2  = (const float*)d_in[4];
  const float* hb12   = (const float*)d_in[5];
  const float* hb13   = (const float*)d_in[6];
  const float* hb23   = (const float*)d_in[7];
  const float* hb1    = (const float*)d_in[8];
  const float* hb2    = (const float*)d_in[9];
  const float* hb3    = (const float*)d_in[10];
  const float* W1     = (const float*)d_in[11];
  const float* b1     = (const float*)d_in[12];
  const float* W2     = (const float*)d_in[13];
  const float* b2     = (const float*)d_in[14];
  const float* projW  = (const float*)d_in[15];
  const float* projb  = (const float*)d_in[16];
  const float* eW1    = (const float*)d_in[17];
  const float* eb1    = (const float*)d_in[18];
  const float* eW2    = (const float*)d_in[19];
  const float* eb2    = (const float*)d_in[20];
  const float* nn_b   = (const float*)d_in[21];
  const float* gWih   = (const float*)d_in[22];
  const float* gWhh   = (const float*)d_in[23];
  const float* gbih   = (const float*)d_in[24];
  const float* gbhh   = (const float*)d_in[25];
  const float* c1W    = (const float*)d_in[26];
  const float* c1b    = (const float*)d_in[27];
  const float* c2W    = (const float*)d_in[28];
  const float* c2b    = (const float*)d_in[29];
  const float* c3W    = (const float*)d_in[30];
  const float* c3b    = (const float*)d_in[31];
  float* out          = (float*)d_out;

  // Workspace layout (floats)
  float* ws = (float*)d_ws;
  size_t off = 0;
  float* W1p   = ws + off; off += 80 * HD;          // 80x128 zero-padded W1
  float* P     = ws + off; off += (size_t)NN * HD;  // ping
  float* Q     = ws + off; off += (size_t)NN * HD;  // pong
  float* nodes = ws + off; off += (size_t)3 * NB * HD;
  float* hdn   = ws + off; off += (size_t)3 * NB * HD;
  float* M2    = ws + off; off += (size_t)HD * 4096;        // permuted eW2
  float* Abuf  = ws + off; off += (size_t)3 * NB * 4096;    // A[v,k,o]
  float* xbuf  = ws + off; off += (size_t)3 * NB * HD;      // x @ eb2
  float* xr    = ws + off; off += (size_t)3 * NB * HD;
  float* WihT  = ws + off; off += (size_t)HD * 384;
  float* WhhT  = ws + off; off += (size_t)HD * 384;
  float* gi    = ws + off; off += (size_t)3 * NB * 384;
  float* gh    = ws + off; off += (size_t)3 * NB * 384;
  float* xnew  = ws + off; off += (size_t)3 * NB * HD;
  float* hgcat = ws + off; off += (size_t)NB * 384;
  float* t1    = ws + off; off += (size_t)NB * HD;
  float* t2    = ws + off; off += (size_t)NB * HD;
  (void)ws_size; (void)n_in; (void)in_sizes; (void)out_size;

  pad_w1<<<ceil_div(80 * HD, 256), 256, 0, stream>>>(W1, W1p);

  // --- three GCN branches, shared weights ---
  const float* hs[3] = {h1, h2, h3};
  for (int t = 0; t < 3; ++t) {
    ring_agg<<<ceil_div(NN * 80, 256), 256, 0, stream>>>(hs[t], P, 74, 80);
    launch_gemm(P, 80, W1p, HD, Q, HD, NN, HD, 80, b1, 1, stream);
    ring_agg<<<ceil_div(NN * HD, 256), 256, 0, stream>>>(Q, P, HD, HD);
    launch_gemm(P, HD, W2, HD, Q, HD, NN, HD, HD, b2, 1, stream);
    pool_scale<<<NB, HD, 0, stream>>>(Q, nodes, solv1, solv2, t);
  }

  // --- projection: hdn = relu(nodes @ proj_W + proj_b) ---
  launch_gemm(nodes, HD, projW, HD, hdn, HD, 3 * NB, HD, HD, projb, 1, stream);

  // --- edge network contraction ---
  ew2_perm<<<ceil_div(32 * HD * HD, 256), 256, 0, stream>>>(eW2, M2);
  // A[v, k*128+o] = sum_h hdn[v,h] * eW2[k,h,o]   (1536 x 128 x 4096 GEMM)
  launch_gemm(hdn, HD, M2, 4096, Abuf, 4096, 3 * NB, 4096, HD, nullptr, 0, stream);
  // xb[v,o] = sum_h hdn[v,h] * eb2[h*128+o]
  launch_gemm(hdn, HD, eb2, HD, xbuf, HD, 3 * NB, HD, HD, nullptr, 0, stream);
  edge_agg<<<3 * NB, HD, 0, stream>>>(Abuf, xbuf, eW1, eb1, nn_b,
                                      hb12, hb13, hb23, hb1, hb2, hb3, xr);

  // --- GRU cell ---
  transpose_k<<<ceil_div(384 * HD, 256), 256, 0, stream>>>(gWih, WihT, 384, HD);
  transpose_k<<<ceil_div(384 * HD, 256), 256, 0, stream>>>(gWhh, WhhT, 384, HD);
  launch_gemm(xr, HD, WihT, 384, gi, 384, 3 * NB, 384, HD, gbih, 0, stream);
  launch_gemm(hdn, HD, WhhT, 384, gh, 384, 3 * NB, 384, HD, gbhh, 0, stream);
  gru_elem<<<ceil_div(3 * NB * HD, 256), 256, 0, stream>>>(gi, gh, hdn, xnew);

  // --- classifier ---
  concat3<<<ceil_div(NB * 384, 256), 256, 0, stream>>>(xnew, hgcat);
  launch_gemm(hgcat, 384, c1W, HD, t1, HD, NB, HD, 384, c1b, 1, stream);
  launch_gemm(t1, HD, c2W, HD, t2, HD, NB, HD, HD, c2b, 1, stream);
  final_out<<<ceil_div(NB * 3, 256), 256, 0, stream>>>(t2, c3W, c3b, out);
}